// LegacySEBlock_14851996909886
// MI455X (gfx1250) — compile-verified
//
#include <hip/hip_runtime.h>
#include <hip/hip_bf16.h>
#include <math.h>

typedef float v2f __attribute__((ext_vector_type(2)));
typedef float v8f __attribute__((ext_vector_type(8)));

#define BB 32
#define CC 256
#define RR 16
#define HW 4096
#define POS_RANK 204   // position = round(0.8*256) = 205 -> 0-based rank 204

// ---------------------------------------------------------------------------
// Kernel 1: y[b,c] = mean over HxW. One 256-thread block per (b,c) plane.
// ---------------------------------------------------------------------------
__global__ __launch_bounds__(256) void se_mean_kernel(const float* __restrict__ x,
                                                      float* __restrict__ y) {
  const float4* p4 = (const float4*)(x + (size_t)blockIdx.x * HW);
  float sum = 0.0f;
  for (int i = threadIdx.x; i < HW / 4; i += 256) {
    float4 v = p4[i];
    sum += (v.x + v.y) + (v.z + v.w);
  }
  // wave32 reduction
  for (int off = 16; off > 0; off >>= 1)
    sum += __shfl_xor(sum, off, 32);
  __shared__ float wsum[8];
  if ((threadIdx.x & 31) == 0) wsum[threadIdx.x >> 5] = sum;
  __syncthreads();
  if (threadIdx.x == 0) {
    float t = 0.0f;
    for (int i = 0; i < 8; ++i) t += wsum[i];
    y[blockIdx.x] = t * (1.0f / (float)HW);
  }
}

// ---------------------------------------------------------------------------
// Kernel 2: full SE core on one workgroup (8 wave32s) using fp32 WMMA.
//   h = relu(y @ W1^T)      : (32x256)x(256x16) -> 2 wave-tiles, K=256
//   s = sigmoid(h @ W2^T)   : (32x16)x(16x256)  -> 32 tiles, K=16
//   thr[b] = 205th-smallest of s[b,:];  mask = (s <= thr)
//
// V_WMMA_F32_16X16X4_F32 operand layout (ISA 7.12.2):
//   A (16x4): lane<16 holds row=lane, K={0,1}; lane>=16 holds row=lane-16, K={2,3}
//   B (4x16): lane<16 holds col=lane, K={0,1}; lane>=16 holds col=lane-16, K={2,3}
//   C/D (16x16): vgpr j, lane<16 -> M=j, N=lane; lane>=16 -> M=j+8, N=lane-16
// ---------------------------------------------------------------------------
__global__ __launch_bounds__(256) void se_core_kernel(const float* __restrict__ y,
                                                      const float* __restrict__ W1,
                                                      const float* __restrict__ W2,
                                                      float* __restrict__ mask) {
  __shared__ float h_s[BB][RR];
  __shared__ float s_s[BB][CC];
  __shared__ float thr[BB];

  const int tid  = threadIdx.x;
  const int wave = tid >> 5;
  const int lane = tid & 31;
  const int l16  = lane & 15;
  const int khalf = (lane >> 4) * 2;   // 0 for lanes 0-15, 2 for lanes 16-31
  const int rofs  = (lane >> 4) * 8;   // C/D row offset for upper half-wave

  // ---- GEMM1: waves 0,1 each produce a 16x16 h-tile (M-tiles of b) ----
  if (wave < 2) {
    const int m0  = wave * 16;
    const int row = m0 + l16;
    v8f acc = {};
    for (int k = 0; k < CC; k += 4) {
      v2f a, b;
      a[0] = y[row * CC + k + khalf];
      a[1] = y[row * CC + k + khalf + 1];
      // B[kk][n] = W1[n][k+kk]  (W1 is R x C row-major)
      b[0] = W1[l16 * CC + k + khalf];
      b[1] = W1[l16 * CC + k + khalf + 1];
      acc = __builtin_amdgcn_wmma_f32_16x16x4_f32(false, a, false, b,
                                                  (short)0, acc, false, false);
    }
    for (int j = 0; j < 8; ++j) {
      float v = acc[j];
      h_s[m0 + j + rofs][l16] = v > 0.0f ? v : 0.0f;   // relu
    }
  }
  __syncthreads();

  // ---- GEMM2: 32 tiles (2 M-tiles x 16 N-tiles), 4 tiles per wave ----
  for (int i = 0; i < 4; ++i) {
    const int t  = wave * 4 + i;
    const int mt = t >> 4;        // 0..1
    const int nt = t & 15;        // 0..15
    const int aRow = mt * 16 + l16;
    const int bCol = nt * 16 + l16;
    v8f acc = {};
    for (int k = 0; k < RR; k += 4) {
      v2f a, b;
      a[0] = h_s[aRow][k + khalf];
      a[1] = h_s[aRow][k + khalf + 1];
      // B[kk][n] = W2[nt*16+n][k+kk]  (W2 is C x R row-major)
      b[0] = W2[bCol * RR + k + khalf];
      b[1] = W2[bCol * RR + k + khalf + 1];
      acc = __builtin_amdgcn_wmma_f32_16x16x4_f32(false, a, false, b,
                                                  (short)0, acc, false, false);
    }
    for (int j = 0; j < 8; ++j) {
      float v = acc[j];
      s_s[mt * 16 + j + rofs][nt * 16 + l16] = 1.0f / (1.0f + __expf(-v));
    }
  }
  __syncthreads();

  // ---- per-row rank selection: wave w handles rows w, w+8, w+16, w+24 ----
  for (int rr = 0; rr < 4; ++rr) {
    const int b = wave + rr * 8;
    float v[8];
    int   rank[8];
    for (int q = 0; q < 8; ++q) { v[q] = s_s[b][lane + 32 * q]; rank[q] = 0; }
    for (int j = 0; j < CC; ++j) {
      const float u = s_s[b][j];            // broadcast read, conflict-free
      for (int q = 0; q < 8; ++q) {
        const int cc = lane + 32 * q;
        rank[q] += ((u < v[q]) || (u == v[q] && j < cc)) ? 1 : 0;
      }
    }
    for (int q = 0; q < 8; ++q)
      if (rank[q] == POS_RANK) thr[b] = v[q];  // exactly one lane matches
  }
  __syncthreads();

  // ---- mask = (s <= thr) as float, matching reference tie semantics ----
  for (int i = tid; i < BB * CC; i += 256) {
    const int b = i >> 8;
    const int c = i & 255;
    mask[i] = (s_s[b][c] <= thr[b]) ? 1.0f : 0.0f;
  }
}

// ---------------------------------------------------------------------------
// Kernel 3: out = x * mask[b,c]. One block per plane; skip read when masked.
// ---------------------------------------------------------------------------
__global__ __launch_bounds__(256) void se_apply_kernel(const float* __restrict__ x,
                                                       const float* __restrict__ mask,
                                                       float* __restrict__ out) {
  const size_t base = (size_t)blockIdx.x * HW;
  const float m = mask[blockIdx.x];
  float4* o4 = (float4*)(out + base);
  if (m == 0.0f) {
    const float4 z = {0.0f, 0.0f, 0.0f, 0.0f};
    for (int i = threadIdx.x; i < HW / 4; i += 256) o4[i] = z;
  } else {
    const float4* x4 = (const float4*)(x + base);
    for (int i = threadIdx.x; i < HW / 4; i += 256) {
      float4 v = x4[i];
      v.x *= m; v.y *= m; v.z *= m; v.w *= m;
      o4[i] = v;
    }
  }
}

// ---------------------------------------------------------------------------
extern "C" void kernel_launch(void* const* d_in, const int* in_sizes, int n_in,
                              void* d_out, int out_size, void* d_ws, size_t ws_size,
                              hipStream_t stream) {
  const float* x  = (const float*)d_in[0];   // (32,256,64,64)
  const float* W1 = (const float*)d_in[1];   // (16,256)
  const float* W2 = (const float*)d_in[2];   // (256,16)
  float* out = (float*)d_out;

  float* y    = (float*)d_ws;                // 8192 f32
  float* mask = y + BB * CC;                 // 8192 f32

  se_mean_kernel<<<BB * CC, 256, 0, stream>>>(x, y);
  se_core_kernel<<<1, 256, 0, stream>>>(y, W1, W2, mask);
  se_apply_kernel<<<BB * CC, 256, 0, stream>>>(x, mask, out);
}